// ManualFeature_3702261809445
// MI455X (gfx1250) — compile-verified
//
#include <hip/hip_runtime.h>

typedef __attribute__((ext_vector_type(2))) float v2f;
typedef __attribute__((ext_vector_type(8))) float v8f;

#define NBINS 17   // buckets 0..15 used, 16 = overflow (>15, discarded)
#define NREP  32   // histogram replicas: tid>>3, limits ds bank serialization

__global__ __launch_bounds__(256) void manual_feature_kernel(
    const float* __restrict__ pcd,   // (B, P, 3)
    const float* __restrict__ locs,  // (L, 3)
    float* __restrict__ out,         // (B, L, 15)
    int P, int L)
{
    __shared__ unsigned int hist[NREP][16][NBINS];

    const int tid  = threadIdx.x;
    const int lane = tid & 31;
    const int wave = tid >> 5;
    const int half = (lane >> 4) & 1;   // 0: lanes 0-15, 1: lanes 16-31
    const int lidx = lane & 15;

    const int ltile = blockIdx.x;       // 16 locations per block
    const int b     = blockIdx.y;

    // ---- zero LDS histogram ----
    for (int e = tid; e < NREP * 16 * NBINS; e += 256)
        ((unsigned int*)hist)[e] = 0u;
    __syncthreads();

    // ---- A matrix: 16 locs x K=4 (xyz + zero pad), pre-scaled by -2 ----
    // f32 16x4 A layout: lanes 0-15 -> (K0,K1) of row M=lane;
    //                    lanes 16-31 -> (K2,K3) of row M=lane-16.
    int lrow   = ltile * 16 + lidx;
    int lclamp = (lrow < L) ? lrow : (L - 1);
    float lx = locs[lclamp * 3 + 0];
    float ly = locs[lclamp * 3 + 1];
    float lz = locs[lclamp * 3 + 2];
    v2f a;
    a.x = half ? (-2.0f * lz) : (-2.0f * lx);
    a.y = half ? 0.0f         : (-2.0f * ly);

    // ---- per-lane squared loc norms for the 8 C rows this lane holds ----
    // C/D layout: lane 0-15 VGPR i -> M=i ; lane 16-31 VGPR i -> M=i+8
    float lnorm[8];
    #pragma unroll
    for (int i = 0; i < 8; ++i) {
        int m  = i + half * 8;
        int gl = ltile * 16 + m;
        int gc = (gl < L) ? gl : (L - 1);
        float x = locs[gc * 3 + 0];
        float y = locs[gc * 3 + 1];
        float z = locs[gc * 3 + 2];
        lnorm[i] = x * x + y * y + z * z;
    }

    const float* pb     = pcd + (size_t)b * P * 3;
    const int    ntiles = (P + 15) >> 4;   // 16 points per tile
    const int    rep    = tid >> 3;        // 32 replicas across the block

    for (int t = wave; t < ntiles; t += 8) {
        int  pidx = t * 16 + lidx;
        bool inb  = (pidx < P);
        int  pc   = inb ? pidx : 0;
        float px = pb[pc * 3 + 0];
        float py = pb[pc * 3 + 1];
        float pz = pb[pc * 3 + 2];
        // padded lanes get a huge coordinate -> overflow bucket -> discarded
        px = inb ? px : 1.0e9f;
        py = inb ? py : 0.0f;
        pz = inb ? pz : 0.0f;
        float pn = px * px + py * py + pz * pz;

        // B matrix 4x16: lanes 0-15 -> (K0,K1)=(x,y) of col N=lane;
        //                lanes 16-31 -> (K2,K3)=(z,0) of col N=lane-16.
        v2f bm;
        bm.x = half ? pz   : px;
        bm.y = half ? 0.0f : py;

        v8f c = {};
        // D = (-2*locs) x pcd^T  ->  c[i] = -2 * (l_m . p_n)
        c = __builtin_amdgcn_wmma_f32_16x16x4_f32(
                /*neg_a=*/false, a, /*neg_b=*/false, bm,
                /*c_mod=*/(short)0, c, /*reuse_a=*/false, /*reuse_b=*/false);

        #pragma unroll
        for (int i = 0; i < 8; ++i) {
            // bucket = ceil(||p-l||) = smallest k with k*k >= d2.
            // Raw v_sqrt_f32 estimate + two exact-square up-corrections:
            // trunc of the ~1ulp estimate lands at most 2 below the true
            // ceiling, and corrections only fire while kf^2 < d2, so the
            // result is exact (squares <= 256 are exactly representable).
            float d2 = c[i] + lnorm[i] + pn;          // ||p - l||^2
            d2 = fminf(fmaxf(d2, 0.0f), 1100.0f);     // clamp: >16 all overflow
            float s  = __builtin_amdgcn_sqrtf(d2);    // single v_sqrt_f32
            float kf = truncf(s);
            kf += (kf * kf < d2) ? 1.0f : 0.0f;
            kf += (kf * kf < d2) ? 1.0f : 0.0f;
            int v = (int)fminf(kf, 16.0f);
            int m = i + half * 8;
            atomicAdd(&hist[rep][m][v], 1u);          // ds_add_u32
        }
    }
    __syncthreads();

    // ---- reduce replicas into replica 0 (one owner thread per (row,bin)) ----
    for (int e = tid; e < 16 * NBINS; e += 256) {
        int row = e / NBINS, bin = e % NBINS;
        unsigned int s = 0;
        #pragma unroll
        for (int r = 0; r < NREP; ++r) s += hist[r][row][bin];
        hist[0][row][bin] = s;
    }
    __syncthreads();

    // ---- cumulative histogram -> output: out[b,l,t] = sum_{v<=t+1} h[v] ----
    for (int e = tid; e < 16 * 15; e += 256) {
        int row = e / 15, tt = e % 15;
        unsigned int cum = 0;
        for (int v = 0; v <= tt + 1; ++v) cum += hist[0][row][v];
        int gl = ltile * 16 + row;
        if (gl < L)
            out[((size_t)b * L + gl) * 15 + tt] = (float)cum;
    }
}

extern "C" void kernel_launch(void* const* d_in, const int* in_sizes, int n_in,
                              void* d_out, int out_size, void* d_ws, size_t ws_size,
                              hipStream_t stream) {
    const float* pcd  = (const float*)d_in[0];   // (B, P, 3) f32
    const float* locs = (const float*)d_in[1];   // (L, 3)    f32
    float*       out  = (float*)d_out;           // (B, L, 15) f32

    int L = in_sizes[1] / 3;
    int B = out_size / (L * 15);
    int P = in_sizes[0] / (B * 3);

    dim3 grid((L + 15) / 16, B);
    manual_feature_kernel<<<grid, 256, 0, stream>>>(pcd, locs, out, P, L);
}